// CKY_39479339385013
// MI455X (gfx1250) — compile-verified
//
#include <hip/hip_runtime.h>
#include <stdint.h>

// ---------------- problem constants (B=16, S=192 from the reference) --------
#define S_DIM   192
#define SP      193            // padded LDS row stride (dwords): 193 % 64 != 0 -> no bank conflicts on column walks
#define BLOCK   1024
#define NWAVES  (BLOCK / 32)
#define NEGBIG  (-1.0e30f)
#define MAXU    6              // ceil((S-1)/32): max per-lane trips in a reduction

// ---------------- CDNA5 async global->LDS prefetch (ASYNCcnt path) ---------
#if defined(__has_builtin)
#if __has_builtin(__builtin_amdgcn_global_load_async_to_lds_b32)
#define USE_ASYNC 1
#endif
#endif
#ifndef USE_ASYNC
#define USE_ASYNC 0
#endif

#if USE_ASYNC
#define AS1 __attribute__((address_space(1)))
#define AS3 __attribute__((address_space(3)))
__device__ __forceinline__ void asyncLoadF32(const float* g, float* l) {
  // builtin prototype takes int* pointers (per hipcc diagnostic)
  __builtin_amdgcn_global_load_async_to_lds_b32(
      (AS1 int*)const_cast<float*>(g), (AS3 int*)l,
      /*offset=*/0, /*cpol=*/0);
}
__device__ __forceinline__ void asyncWait0() {
  asm volatile("s_wait_asynccnt 0" ::: "memory");
}
#else
__device__ __forceinline__ void asyncLoadF32(const float* g, float* l) { *l = *g; }
__device__ __forceinline__ void asyncWait0() {}
#endif

// ---------------- math helpers ----------------------------------------------
__device__ __forceinline__ float softplusf(float x) {
  // matches jax.nn.softplus: max(x,0) + log1p(exp(-|x|))
  return fmaxf(x, 0.0f) + log1pf(expf(-fabsf(x)));
}
__device__ __forceinline__ float sigmoidf_(float x) {
  if (x >= 0.0f) return 1.0f / (1.0f + expf(-x));
  float e = expf(x);
  return e / (1.0f + e);
}
__device__ __forceinline__ float waveMax(float v) {
#pragma unroll
  for (int off = 16; off > 0; off >>= 1) v = fmaxf(v, __shfl_xor(v, off, 32));
  return v;
}
__device__ __forceinline__ float waveSum(float v) {
#pragma unroll
  for (int off = 16; off > 0; off >>= 1) v += __shfl_xor(v, off, 32);
  return v;
}

// ---------------- kernel: one workgroup per batch element -------------------
// LDS layout (dynamic, ~298 KB of the 320 KB WGP LDS):
//   chart[S*SP]  : log inside scores (padded rows)
//   Glog [S*SP]  : log g + sp - chart (backward adjoint, padded rows)
//   stage[2*S]   : double-buffered async staging of the next rules diagonal
__global__ __launch_bounds__(BLOCK)
void cky_inside_kernel(const float* __restrict__ span,
                       const int*   __restrict__ lens,
                       float* __restrict__ outZ,
                       float* __restrict__ outM) {
  const int b    = blockIdx.x;
  const int S    = S_DIM;
  const int tid  = threadIdx.x;
  const int wave = tid >> 5;
  const int lane = tid & 31;

  const float* rules = span + (size_t)b * S * S;
  float*       marg  = outM + (size_t)b * S * S;
  const int L = lens[b];

  extern __shared__ float lds[];
  float* chart = lds;                 // S*SP
  float* Glog  = lds + S * SP;        // S*SP
  float* stage = lds + 2 * S * SP;    // 2*S

  // ---- init: chart diagonal, Glog = -inf, marginal slice = 0 ----
  // (cells with j >= L never influence Z or the marginals, so the sweeps
  //  below are truncated to widths < L; marginal there stays 0.)
  for (int idx = tid; idx < S * SP; idx += BLOCK) Glog[idx] = NEGBIG;
  for (int idx = tid; idx < S * S;  idx += BLOCK) marg[idx] = 0.0f;
  for (int i = tid; i < S; i += BLOCK)
    chart[i * SP + i] = softplusf(rules[i * S + i]);
  // prefetch rules diagonal for width 1
  if (tid < L - 1)
    asyncLoadF32(&rules[tid * S + tid + 1], &stage[(1 & 1) * S + tid]);

  // =========================== forward ======================================
  for (int w = 1; w < L; ++w) {
    asyncWait0();        // this wave's staged diagonal (width w) has landed
    __syncthreads();     // all staging + all width-(w-1) chart writes visible

    const float* cur = &stage[(w & 1) * S];
    // kick off async prefetch of the next diagonal (overlaps with compute)
    if (w + 1 < L) {
      const int nn = L - (w + 1);
      if (tid < nn)
        asyncLoadF32(&rules[tid * S + tid + w + 1],
                     &stage[((w + 1) & 1) * S + tid]);
    }

    const int nc = L - w;              // only cells with j < L matter
    const int nU = (w + 31) >> 5;      // uniform per-lane trip count (<= MAXU)
    for (int c = wave; c < nc; c += NWAVES) {
      const int i = c, j = c + w;
      const float* rowL = &chart[i * SP];
      float vals[MAXU];
      float m = NEGBIG;
#pragma unroll
      for (int u = 0; u < MAXU; ++u) {
        if (u < nU) {
          const int k = i + lane + (u << 5);
          const float t =
              (k < j) ? rowL[k] + chart[(k + 1) * SP + j] : NEGBIG;
          vals[u] = t;
          m = fmaxf(m, t);
        }
      }
      m = waveMax(m);
      float s = 0.0f;
#pragma unroll
      for (int u = 0; u < MAXU; ++u) {
        if (u < nU) s += expf(vals[u] - m);   // NEGBIG entries underflow to 0
      }
      s = waveSum(s);
      if (lane == 0)
        chart[i * SP + j] = softplusf(cur[c]) + m + logf(s);
    }
    // no trailing barrier: top-of-loop barrier covers it
  }
  __syncthreads();

  if (tid == 0) outZ[b] = chart[0 * SP + (L - 1)];
  __syncthreads();

  // =========================== backward =====================================
  // g[a,b] = sum_{j>b} exp(Glog[a,j] + chart[a,b] + chart[b+1,j])
  //        + sum_{i<a} exp(Glog[i,b] + chart[i,a-1] + chart[a,b])
  // (+1 at the root (0, L-1)); marginal = g * sigmoid(rules);
  // Glog[a,b] = log g + softplus(rules) - chart[a,b].
  for (int w = L - 1; w >= 0; --w) {
    const int nc = L - w;
    for (int c = wave; c < nc; c += NWAVES) {
      const int a = c, bb = c + w;
      const float cab = chart[a * SP + bb];
      float s = 0.0f;
      // left-child-of parents (a, j), j in (bb, L-1]
      {
        const float* gRow = &Glog[a * SP];
        const float* cRow = &chart[(bb + 1) * SP];
        for (int j = bb + 1 + lane; j < L; j += 32)
          s += expf(gRow[j] + cab + cRow[j]);
      }
      // right-child-of parents (i, bb), i in [0, a-1]
      for (int i = lane; i < a; i += 32)
        s += expf(Glog[i * SP + bb] + chart[i * SP + (a - 1)] + cab);
      s = waveSum(s);
      if (a == 0 && bb == L - 1) s += 1.0f;  // seed dZ/dchart at the root
      if (lane == 0) {
        const float r = rules[a * S + bb];
        marg[a * S + bb] = s * sigmoidf_(r);
        Glog[a * SP + bb] = logf(s) + softplusf(r) - cab;
      }
    }
    __syncthreads();
  }
}

// ---------------- host launch ------------------------------------------------
extern "C" void kernel_launch(void* const* d_in, const int* in_sizes, int n_in,
                              void* d_out, int out_size, void* d_ws, size_t ws_size,
                              hipStream_t stream) {
  (void)n_in; (void)out_size; (void)d_ws; (void)ws_size;
  const float* span = (const float*)d_in[0];
  const int*   lens = (const int*)d_in[1];
  const int B = in_sizes[1];            // sequence_lengths has B elements

  float* outZ = (float*)d_out;          // Z: B floats
  float* outM = outZ + B;               // marginal: B*S*S floats

  const size_t shbytes =
      (size_t)(2 * S_DIM * SP + 2 * S_DIM) * sizeof(float);  // ~298 KB (fits 320 KB WGP LDS)

  cky_inside_kernel<<<dim3(B), dim3(BLOCK), shbytes, stream>>>(span, lens, outZ, outM);
}